// GruNet_60782377173235
// MI455X (gfx1250) — compile-verified
//
#include <hip/hip_runtime.h>
#include <hip/hip_bf16.h>
#include <math.h>

// ---------------------------------------------------------------------------
// GRU rollout network on gfx1250 (MI455X), full fp32 via V_WMMA_F32_16X16X4_F32
//
// Sizes (fixed by the reference): T=128, N=64, H=1024, 3H=3072.
//   gi  = x(8192x1024) @ w_ih^T(1024x3072) + b_ih          -> big parallel GEMM
//   scan: 128 x { gh = (h*m_t)(64x1024) @ w_hh^T + b_hh ;  gates -> h, ys[t] }
//   out = relu(ys(8192x1024) @ w_out^T(1024x1024) + b_out)
//
// Roofline: ~120 GFLOP fp32 total; all operands (~190 MB) are L2-resident,
// HBM traffic ~100 MB (~4.3us @ 23.3 TB/s) -> matrix-unit + scan-latency
// bound. K-chunk staging is double-buffered through LDS using the CDNA5
// async global->LDS DMA path (ASYNCcnt) so chunk k+1 is in flight during
// chunk k's WMMA chain; falls back to a sched_barrier-pinned register
// pipeline if the async builtins are unavailable.
// ---------------------------------------------------------------------------

typedef __attribute__((ext_vector_type(2))) float v2f;
typedef __attribute__((ext_vector_type(8))) float v8f;
typedef int v4i_vs __attribute__((vector_size(16)));   // matches builtin param

#define TT 128
#define NENV 64
#define HID 1024
#define H3 3072

#if __has_builtin(__builtin_amdgcn_global_load_async_to_lds_b128) && \
    __has_builtin(__builtin_amdgcn_s_wait_asynccnt)
#define USE_ASYNC_LDS 1
#else
#define USE_ASYNC_LDS 0
#endif

// Exactly-typed address-space casts for the async-DMA builtin:
//   global source: AS1 pointer to int4 (64-bit); bit pattern == generic.
//   LDS dest:      AS3 pointer to int4 (32-bit); generic LDS pointer's low
//                  32 bits are the wave-relative LDS address (flat aperture).
#define AS1_PTR(p) ((__attribute__((address_space(1))) v4i_vs*)(uintptr_t)(p))
#define AS3_PTR(p) ((__attribute__((address_space(3))) v4i_vs*)(unsigned)(uintptr_t)(p))

// ---------------------------------------------------------------------------
// Generic C = A(MxK) * W^T (W is NxK row-major) + bias(N), optional ReLU,
// optional per-row mask: A row m scaled by (1 - done_row[m]), applied to the
// A fragment (per-lane scalar) right before each WMMA -- exact in fp32.
//
// Block: 256 threads = 8 waves as 2(M-tiles) x 4(N-tiles); block tile 32x64.
// K chunks of 64 staged into double-buffered LDS (row stride 68 floats ->
// conflict-free b64 fragment reads).
//
// WMMA fragment layout per CDNA5 ISA 7.12.2:
//   A 16x4 : lanes 0-15 row M=lane  (K=0,1 in .x,.y); lanes 16-31 K=2,3
//   B 4x16 : lanes 0-15 col N=lane  (K=0,1)         ; lanes 16-31 K=2,3
//   C 16x16: VGPR i -> row i (lanes 0-15) / row i+8 (lanes 16-31), col = lane&15
// ---------------------------------------------------------------------------
template <bool RELU, bool MASK>
__global__ __launch_bounds__(256) void gemm_wmma_f32(
    const float* __restrict__ A,       // M x K, row-major
    const float* __restrict__ W,       // N x K, row-major (we compute A @ W^T)
    const float* __restrict__ bias,    // N
    const float* __restrict__ done_row,// M entries (only if MASK)
    float* __restrict__ C,             // M x N
    int M, int Nn, int K)
{
    constexpr int KC  = 64;   // K chunk
    constexpr int SAS = 68;   // padded LDS row stride (floats)

    __shared__ float As[2][32 * SAS];   // 2 x 8.5 KB
    __shared__ float Bs[2][64 * SAS];   // 2 x 17 KB

    const int tid  = threadIdx.x;
    const int lane = tid & 31;
    const int wave = tid >> 5;
    const int wm   = wave >> 2;      // 0..1  (M tile within block)
    const int wn   = wave & 3;       // 0..3  (N tile within block)
    const int lo   = lane & 15;
    const int hi   = lane >> 4;      // 0/1 : which K-pair half

    const int m0 = blockIdx.x * 32;
    const int n0 = blockIdx.y * 64;

    // Per-thread staging coordinates are k-invariant.
    // A: 2 x b128 / thread (32 rows x 64 K); W: 4 x b128 / thread (64 x 64).
    int aoff[2];
    const float* aptr[2];
#pragma unroll
    for (int j = 0; j < 2; ++j) {
        const int idx = tid + j * 256;          // 0..511
        const int row = idx >> 4;               // 0..31
        const int kq  = (idx & 15) << 2;        // 0..60
        aoff[j] = row * SAS + kq;
        aptr[j] = A + (size_t)(m0 + row) * K + kq;
    }
    int boff[4];
    const float* bptr[4];
#pragma unroll
    for (int j = 0; j < 4; ++j) {
        const int idx = tid + j * 256;          // 0..1023
        const int row = idx >> 4;               // 0..63
        const int kq  = (idx & 15) << 2;
        boff[j] = row * SAS + kq;
        bptr[j] = W + (size_t)(n0 + row) * K + kq;
    }

    // Episode-reset mask as a per-lane A-fragment scale (rows of this wave's
    // 16x16 tile: M = m0 + wm*16 + lo for both lane halves).
    const float fragMask = MASK ? (1.0f - done_row[m0 + wm * 16 + lo]) : 1.0f;

    // Accumulator starts at bias[n] (same value in all 8 row-VGPRs).
    const float bval = bias[n0 + wn * 16 + lo];
    v8f c;
#pragma unroll
    for (int i = 0; i < 8; ++i) c[i] = bval;

    const int afrag = (wm * 16 + lo) * SAS + (hi << 1);
    const int bfrag = (wn * 16 + lo) * SAS + (hi << 1);

    // 16 x v_wmma_f32_16x16x4_f32 over one resident K-chunk.
    auto mma_chunk = [&](const float* __restrict__ as,
                         const float* __restrict__ bs) {
#pragma unroll
        for (int kk = 0; kk < KC; kk += 4) {
            v2f a = *(const v2f*)(as + afrag + kk);
            v2f b = *(const v2f*)(bs + bfrag + kk);
            if (MASK) { a.x *= fragMask; a.y *= fragMask; }
            c = __builtin_amdgcn_wmma_f32_16x16x4_f32(
                    false, a, false, b, (short)0, c, false, false);
        }
    };

#if USE_ASYNC_LDS
    // ---- CDNA5 async global->LDS DMA, double-buffered ----
    auto issue_chunk = [&](int p, int k0) {
#pragma unroll
        for (int j = 0; j < 2; ++j)
            __builtin_amdgcn_global_load_async_to_lds_b128(
                AS1_PTR(aptr[j] + k0), AS3_PTR(&As[p][aoff[j]]), 0, 0);
#pragma unroll
        for (int j = 0; j < 4; ++j)
            __builtin_amdgcn_global_load_async_to_lds_b128(
                AS1_PTR(bptr[j] + k0), AS3_PTR(&Bs[p][boff[j]]), 0, 0);
    };

    int p = 0;
    issue_chunk(0, 0);
    __builtin_amdgcn_s_wait_asynccnt(0);
    __syncthreads();

    for (int k0 = KC; k0 < K; k0 += KC) {
        issue_chunk(p ^ 1, k0);          // DMA next chunk (in flight below)
        mma_chunk(As[p], Bs[p]);         // compute resident chunk
        __builtin_amdgcn_s_wait_asynccnt(0);
        __syncthreads();
        p ^= 1;
    }
    mma_chunk(As[p], Bs[p]);
#else
    // ---- fallback: register-pipelined staging, schedule pinned ----
    float4 aReg[2], bReg[4];
    auto fetch_chunk = [&](int k0) {
#pragma unroll
        for (int j = 0; j < 2; ++j) aReg[j] = *(const float4*)(aptr[j] + k0);
#pragma unroll
        for (int j = 0; j < 4; ++j) bReg[j] = *(const float4*)(bptr[j] + k0);
    };
    auto commit_chunk = [&]() {
#pragma unroll
        for (int j = 0; j < 2; ++j) *(float4*)(&As[0][aoff[j]]) = aReg[j];
#pragma unroll
        for (int j = 0; j < 4; ++j) *(float4*)(&Bs[0][boff[j]]) = bReg[j];
    };

    fetch_chunk(0);
    commit_chunk();
    __syncthreads();
    for (int k0 = KC; k0 < K; k0 += KC) {
        fetch_chunk(k0);                 // loads in flight during WMMAs
#if __has_builtin(__builtin_amdgcn_sched_barrier)
        __builtin_amdgcn_sched_barrier(0);  // forbid sinking loads below MMAs
#endif
        mma_chunk(As[0], Bs[0]);
        __syncthreads();
        commit_chunk();
        __syncthreads();
    }
    mma_chunk(As[0], Bs[0]);
#endif

    // ---- store C tile ----
    const int ccol = n0 + wn * 16 + lo;
#pragma unroll
    for (int i = 0; i < 8; ++i) {
        float v = c[i];
        if (RELU) v = fmaxf(v, 0.0f);
        const int crow = m0 + wm * 16 + i + (hi << 3);
        C[(size_t)crow * Nn + ccol] = v;
    }
}

// ---------------------------------------------------------------------------
// Per-step GRU gates (elementwise, 64x1024):
//   m  = 1 - done[t,n] ; hm = h*m  (gh was computed from masked h already)
//   r  = sigmoid(gi_r + gh_r); z = sigmoid(gi_z + gh_z)
//   nn = tanh(gi_n + r*gh_n);  h' = (1-z)*nn + z*hm
// ---------------------------------------------------------------------------
__global__ __launch_bounds__(256) void gru_gate(
    const float* __restrict__ gi_t,   // 64 x 3072 (this step's slice)
    const float* __restrict__ gh,     // 64 x 3072
    const float* __restrict__ done_t, // 64
    float* __restrict__ h,            // 64 x 1024 (in/out)
    float* __restrict__ ys_t)         // 64 x 1024
{
    const int idx = blockIdx.x * 256 + threadIdx.x;   // 0 .. 65535
    const int n = idx >> 10;
    const int j = idx & 1023;

    const float* gin = gi_t + (size_t)n * H3;
    const float* ghn = gh   + (size_t)n * H3;

    const float m  = 1.0f - done_t[n];
    const float hm = h[idx] * m;

    const float r  = 1.0f / (1.0f + __expf(-(gin[j]            + ghn[j])));
    const float z  = 1.0f / (1.0f + __expf(-(gin[HID + j]      + ghn[HID + j])));
    const float nn = tanhf(gin[2 * HID + j] + r * ghn[2 * HID + j]);

    const float hnew = (1.0f - z) * nn + z * hm;
    h[idx]    = hnew;
    ys_t[idx] = hnew;
}

// ---------------------------------------------------------------------------
extern "C" void kernel_launch(void* const* d_in, const int* in_sizes, int n_in,
                              void* d_out, int out_size, void* d_ws, size_t ws_size,
                              hipStream_t stream) {
    const float* x     = (const float*)d_in[0];   // (T*N, H)
    const float* hxs   = (const float*)d_in[1];   // (N, H)
    const float* done  = (const float*)d_in[2];   // (T*N, 1)
    const float* w_ih  = (const float*)d_in[3];   // (3H, H)
    const float* w_hh  = (const float*)d_in[4];   // (3H, H)
    const float* b_ih  = (const float*)d_in[5];   // (3H)
    const float* b_hh  = (const float*)d_in[6];   // (3H)
    const float* w_out = (const float*)d_in[7];   // (H, H)
    const float* b_out = (const float*)d_in[8];   // (H)

    float* out   = (float*)d_out;                       // (T*N, H)
    float* hlast = out + (size_t)TT * NENV * HID;       // (N, H)

    // Workspace layout (all fp32): gi | ys | gh | h   (~129 MB)
    float* gi = (float*)d_ws;                            // T*N x 3H
    float* ys = gi + (size_t)TT * NENV * H3;             // T*N x H
    float* gh = ys + (size_t)TT * NENV * HID;            // N x 3H
    float* h  = gh + (size_t)NENV * H3;                  // N x H

    // h <- hxs (fresh every call: h is mutated during the scan)
    (void)hipMemcpyAsync(h, hxs, (size_t)NENV * HID * sizeof(float),
                         hipMemcpyDeviceToDevice, stream);

    // gi = x @ w_ih^T + b_ih   (M=8192, N=3072, K=1024)
    gemm_wmma_f32<false, false>
        <<<dim3((TT * NENV) / 32, H3 / 64), 256, 0, stream>>>(
            x, w_ih, b_ih, nullptr, gi, TT * NENV, H3, HID);

    // Sequential scan: 128 dependent (64x3072x1024) GEMMs + gate kernels.
    for (int t = 0; t < TT; ++t) {
        const float* done_t = done + (size_t)t * NENV;
        // gh = (h * m_t) @ w_hh^T + b_hh   (mask applied to A fragments)
        gemm_wmma_f32<false, true>
            <<<dim3(NENV / 32, H3 / 64), 256, 0, stream>>>(
                h, w_hh, b_hh, done_t, gh, NENV, H3, HID);
        gru_gate<<<(NENV * HID) / 256, 256, 0, stream>>>(
            gi + (size_t)t * NENV * H3, gh, done_t, h,
            ys + (size_t)t * NENV * HID);
    }

    // out = relu(ys @ w_out^T + b_out)   (M=8192, N=1024, K=1024)
    gemm_wmma_f32<true, false>
        <<<dim3((TT * NENV) / 32, HID / 64), 256, 0, stream>>>(
            ys, w_out, b_out, nullptr, out, TT * NENV, HID, HID);

    // h_last
    (void)hipMemcpyAsync(hlast, h, (size_t)NENV * HID * sizeof(float),
                         hipMemcpyDeviceToDevice, stream);
}